// NeuConHead_9887014716206
// MI455X (gfx1250) — compile-verified
//
#include <hip/hip_runtime.h>
#include <hip/hip_bf16.h>

#define VOXEL 0.04f
#define MPRE  32768
#define NVOX  (MPRE * 8)      // 262144
#define NVIEW 9
#define CCH   24
#define IH    120
#define IW    160
#define HWIMG (IH * IW)
#define CPRE  50
#define KDIM  75              // 24 feats + 1 zn + 50 pre_feat
#define KPAD  76              // pad to multiple of 4
#define NPAD  32              // 24 -> two 16-col WMMA panels

// workspace layout (floats):
//   [0] sum_z  [1] n_pos  [2] sum_sq  [3] zmean  [4] znorm
//   [4096 ..)                features  N*24
//   [4096 + N*24 ..)         z         N
#define WS_FEAT 4096
#define WS_Z    (WS_FEAT + NVOX * CCH)

typedef float v2f __attribute__((ext_vector_type(2)));
typedef float v8f __attribute__((ext_vector_type(8)));

__device__ __constant__ int c_offx[8] = {0, 1, 0, 0, 1, 1, 0, 1};
__device__ __constant__ int c_offy[8] = {0, 0, 1, 0, 1, 0, 1, 1};
__device__ __constant__ int c_offz[8] = {0, 0, 0, 1, 0, 1, 1, 1};

__global__ void k_init(float* ws) {
    if (threadIdx.x < 8) ws[threadIdx.x] = 0.0f;
}

// -------- back-projection + r_coords + count + (sum_z, n_pos) partials -----
__global__ __launch_bounds__(256) void k_project(
    const int* __restrict__ pre_coords, const float* __restrict__ feats,
    const float* __restrict__ KRcam, const float* __restrict__ origin,
    const float* __restrict__ w2ac, float* __restrict__ out,
    float* __restrict__ ws) {
    const int g   = blockIdx.x * 256 + threadIdx.x;   // exact fit: 1024*256 = NVOX
    const int pre = g >> 3;
    const int oi  = g & 7;

    const int xi = pre_coords[pre * 4 + 1] + c_offx[oi];
    const int yi = pre_coords[pre * 4 + 2] + c_offy[oi];
    const int zi = pre_coords[pre * 4 + 3] + c_offz[oi];

    const float wx = (float)xi * VOXEL + origin[0];
    const float wy = (float)yi * VOXEL + origin[1];
    const float wz = (float)zi * VOXEL + origin[2];

    // r_coords = [w2ac[0:3,:] @ homog , batch]
    #pragma unroll
    for (int i = 0; i < 3; ++i) {
        float c = w2ac[i * 4 + 0] * wx + w2ac[i * 4 + 1] * wy +
                  w2ac[i * 4 + 2] * wz + w2ac[i * 4 + 3];
        out[NVOX * 3 + 4 * g + i] = c;
    }
    out[NVOX * 3 + 4 * g + 3] = (float)pre_coords[pre * 4 + 0];

    float fsum[CCH];
    #pragma unroll
    for (int c = 0; c < CCH; ++c) fsum[c] = 0.0f;
    float zsum = 0.0f, cnt = 0.0f;

    for (int v = 0; v < NVIEW; ++v) {
        const float* P = KRcam + v * 16;
        const float ix = P[0] * wx + P[1] * wy + P[2]  * wz + P[3];
        const float iy = P[4] * wx + P[5] * wy + P[6]  * wz + P[7];
        const float iz = P[8] * wx + P[9] * wy + P[10] * wz + P[11];
        const float sz = (fabsf(iz) > 1e-9f) ? iz : 1e-9f;
        const float px = ix / sz;
        const float py = iy / sz;
        const bool msk = (px >= 0.0f) && (px <= (float)(IW - 1)) &&
                         (py >= 0.0f) && (py <= (float)(IH - 1)) && (iz > 0.0f);
        if (!msk) continue;
        cnt  += 1.0f;
        zsum += iz;

        const float fx0 = floorf(px), fy0 = floorf(py);
        const int x0 = (int)fx0, y0 = (int)fy0;
        const float ax = px - fx0, ay = py - fy0;

        int   xs[4] = {x0, x0 + 1, x0, x0 + 1};
        int   ys[4] = {y0, y0, y0 + 1, y0 + 1};
        float wt[4] = {(1.f - ax) * (1.f - ay), ax * (1.f - ay),
                       (1.f - ax) * ay,         ax * ay};
        int   ad[4];
        #pragma unroll
        for (int j = 0; j < 4; ++j) {
            const bool ok = (xs[j] >= 0) && (xs[j] < IW) && (ys[j] >= 0) && (ys[j] < IH);
            const int xc = min(max(xs[j], 0), IW - 1);
            const int yc = min(max(ys[j], 0), IH - 1);
            ad[j] = yc * IW + xc;
            wt[j] = ok ? wt[j] : 0.0f;
        }
        const float* img = feats + (size_t)v * CCH * HWIMG;
        #pragma unroll
        for (int c = 0; c < CCH; ++c) {
            const float* ic = img + c * HWIMG;
            fsum[c] += wt[0] * ic[ad[0]] + wt[1] * ic[ad[1]] +
                       wt[2] * ic[ad[2]] + wt[3] * ic[ad[3]];
        }
    }

    const float inv = 1.0f / fmaxf(cnt, 1.0f);
    #pragma unroll
    for (int c = 0; c < CCH; ++c) ws[WS_FEAT + (size_t)g * CCH + c] = fsum[c] * inv;
    const float z = zsum * inv;
    ws[WS_Z + g]    = z;
    out[NVOX * 2 + g] = cnt;

    // block reduce (z if z>0, pos count) -> atomics
    __shared__ float s_z[256], s_c[256];
    s_z[threadIdx.x] = (z > 0.0f) ? z : 0.0f;
    s_c[threadIdx.x] = (z > 0.0f) ? 1.0f : 0.0f;
    __syncthreads();
    for (int s = 128; s > 0; s >>= 1) {
        if (threadIdx.x < s) {
            s_z[threadIdx.x] += s_z[threadIdx.x + s];
            s_c[threadIdx.x] += s_c[threadIdx.x + s];
        }
        __syncthreads();
    }
    if (threadIdx.x == 0) {
        atomicAdd(&ws[0], s_z[0]);
        atomicAdd(&ws[1], s_c[0]);
    }
}

__global__ void k_zmean(float* ws) { ws[3] = ws[0] / fmaxf(ws[1], 1.0f); }

__global__ __launch_bounds__(256) void k_sumsq(float* __restrict__ ws) {
    const int g = blockIdx.x * 256 + threadIdx.x;
    const float zmean = ws[3];
    const float z = ws[WS_Z + g];
    const float d = (z > 0.0f) ? (z - zmean) : 0.0f;
    __shared__ float s[256];
    s[threadIdx.x] = d * d;
    __syncthreads();
    for (int st = 128; st > 0; st >>= 1) {
        if (threadIdx.x < st) s[threadIdx.x] += s[threadIdx.x + st];
        __syncthreads();
    }
    if (threadIdx.x == 0) atomicAdd(&ws[2], s[0]);
}

__global__ void k_znorm(float* ws) { ws[4] = sqrtf(ws[2]) + 1e-5f; }

// -------- MLP head: h = relu(feat @ W_sp + b) via V_WMMA_F32_16X16X4_F32 ----
__global__ __launch_bounds__(256) void k_mlp(
    const float* __restrict__ pre_feat, const float* __restrict__ W_sp,
    const float* __restrict__ b_sp, const float* __restrict__ W_t,
    const float* __restrict__ b_t, const float* __restrict__ W_o,
    const float* __restrict__ b_o, float* __restrict__ out,
    const float* __restrict__ ws) {
    __shared__ float wlds[KPAD * NPAD];        // zero-padded W_sp, K-major
    __shared__ float flds[8][16 * KPAD];       // per-wave 16x76 feat tile
    __shared__ float hlds[8][16 * 25];         // per-wave 16x24 h tile (pad 25)

    const int lane = threadIdx.x & 31;
    const int wave = threadIdx.x >> 5;
    const int half = lane >> 4;                // 0: lanes 0-15, 1: lanes 16-31
    const int m    = lane & 15;

    // stage padded weights (shared by all 8 waves)
    for (int i = threadIdx.x; i < KPAD * NPAD; i += 256) {
        const int k = i >> 5, n = i & 31;
        wlds[i] = (k < KDIM && n < CCH) ? W_sp[k * CCH + n] : 0.0f;
    }

    const int tile = blockIdx.x * 8 + wave;
    const int g0   = tile * 16;
    const float zmean = ws[3];
    const float zninv = 1.0f / ws[4];

    // build feat tile: [0..23]=features, [24]=zn, [25..74]=pre_feat, [75]=0
    for (int i = lane; i < 16 * KPAD; i += 32) {
        const int row = i / KPAD;
        const int k   = i - row * KPAD;
        const int g   = g0 + row;
        float v;
        if (k < CCH) {
            v = ws[WS_FEAT + (size_t)g * CCH + k];
        } else if (k == CCH) {
            const float z = ws[WS_Z + g];
            v = (z > 0.0f) ? (z - zmean) * zninv : 0.0f;
        } else if (k < KDIM) {
            v = pre_feat[(size_t)(g >> 3) * CPRE + (k - (CCH + 1))];
        } else {
            v = 0.0f;
        }
        flds[wave][i] = v;
    }
    __syncthreads();

    v8f acc0 = {0.f, 0.f, 0.f, 0.f, 0.f, 0.f, 0.f, 0.f};
    v8f acc1 = {0.f, 0.f, 0.f, 0.f, 0.f, 0.f, 0.f, 0.f};
    #pragma unroll
    for (int kk = 0; kk < KPAD / 4; ++kk) {
        const int klo = kk * 4 + 2 * half;     // lanes 0-15: K+0/1, 16-31: K+2/3
        v2f a, b0, b1;
        a.x  = flds[wave][m * KPAD + klo];
        a.y  = flds[wave][m * KPAD + klo + 1];
        b0.x = wlds[klo * NPAD + m];
        b0.y = wlds[(klo + 1) * NPAD + m];
        b1.x = wlds[klo * NPAD + 16 + m];
        b1.y = wlds[(klo + 1) * NPAD + 16 + m];
        acc0 = __builtin_amdgcn_wmma_f32_16x16x4_f32(false, a, false, b0,
                                                     (short)0, acc0, false, false);
        acc1 = __builtin_amdgcn_wmma_f32_16x16x4_f32(false, a, false, b1,
                                                     (short)0, acc1, false, false);
    }

    // bias + relu, spill to LDS in C/D layout: VGPR r -> M = r + 8*half, N = m
    const float bs0 = b_sp[m];
    const float bs1 = (m < 8) ? b_sp[16 + m] : 0.0f;
    #pragma unroll
    for (int r = 0; r < 8; ++r) {
        const int row = r + 8 * half;
        hlds[wave][row * 25 + m] = fmaxf(acc0[r] + bs0, 0.0f);
        if (m < 8) hlds[wave][row * 25 + 16 + m] = fmaxf(acc1[r] + bs1, 0.0f);
    }
    __syncthreads();

    // lanes 0-15: tsdf for row m; lanes 16-31: occ for row m
    const float* wv = half ? W_o : W_t;
    float acc = half ? b_o[0] : b_t[0];
    #pragma unroll
    for (int n = 0; n < CCH; ++n) acc += hlds[wave][m * 25 + n] * wv[n];
    out[2 * (g0 + m) + half] = acc;
}

extern "C" void kernel_launch(void* const* d_in, const int* in_sizes, int n_in,
                              void* d_out, int out_size, void* d_ws, size_t ws_size,
                              hipStream_t stream) {
    const float* pre_feat   = (const float*)d_in[0];
    const int*   pre_coords = (const int*)  d_in[1];
    const float* feats      = (const float*)d_in[2];
    const float* KRcam      = (const float*)d_in[3];
    const float* origin     = (const float*)d_in[4];
    const float* w2ac       = (const float*)d_in[5];
    const float* W_sp       = (const float*)d_in[6];
    const float* b_sp       = (const float*)d_in[7];
    const float* W_t        = (const float*)d_in[8];
    const float* b_t        = (const float*)d_in[9];
    const float* W_o        = (const float*)d_in[10];
    const float* b_o        = (const float*)d_in[11];
    float* out = (float*)d_out;
    float* ws  = (float*)d_ws;

    k_init<<<1, 32, 0, stream>>>(ws);
    k_project<<<NVOX / 256, 256, 0, stream>>>(pre_coords, feats, KRcam, origin,
                                              w2ac, out, ws);
    k_zmean<<<1, 1, 0, stream>>>(ws);
    k_sumsq<<<NVOX / 256, 256, 0, stream>>>(ws);
    k_znorm<<<1, 1, 0, stream>>>(ws);
    k_mlp<<<NVOX / (16 * 8), 256, 0, stream>>>(pre_feat, W_sp, b_sp, W_t, b_t,
                                               W_o, b_o, out, ws);
}